// Transform_82008105550484
// MI455X (gfx1250) — compile-verified
//
#include <hip/hip_runtime.h>

typedef float v2f __attribute__((ext_vector_type(2)));
typedef float v8f __attribute__((ext_vector_type(8)));

#define FLAME_PI  3.14159265358979323846f
#define FLAME_EPS 1e-6f
#define WPB       8   // waves per 256-thread block

struct Mat3 { float m[9]; };   // row-major 3x3, register-resident
struct W8   { float w[8]; };   // variation weights, register-resident

// 32-bit LDS byte offset of a __shared__ object (addrspacecast generic->local,
// then ptrtoint). Also escapes the LDS address into the asm operands below so
// the compiler must treat the async instructions as writers of the tiles.
__device__ __forceinline__ unsigned lds_off_u32(void* p) {
    return (unsigned)(unsigned long long)(__attribute__((address_space(3))) void*)p;
}

// Compute routed (fx, fy, z) for one point. All selection is branchless over
// REGISTER values: with uniformly random func_choices every wave contains all
// 8 variations, so v_cndmask chains beat both divergent branches and a
// dependent gathered load of var_params[v].
__device__ __forceinline__ void flame_core(
    float px, float py, float pz, int v,
    const Mat3& L, const W8& W,
    float& fx_o, float& fy_o, float& z_o)
{
    // p = point @ linear
    float x = fmaf(px, L.m[0], fmaf(py, L.m[3], pz * L.m[6]));
    float y = fmaf(px, L.m[1], fmaf(py, L.m[4], pz * L.m[7]));
    float z = fmaf(px, L.m[2], fmaf(py, L.m[5], pz * L.m[8]));

    float r2     = fmaf(x, x, y * y);
    float r      = sqrtf(r2 + FLAME_EPS);
    float theta  = atan2f(x, y);               // NOTE: jnp.arctan2(x, y) order
    float inv_r2 = 1.0f / (r2 + FLAME_EPS);
    float inv_r  = 1.0f / r;
    float sr2    = sinf(r2), cr2 = cosf(r2);
    float sx     = sinf(x),  sy  = sinf(y);
    float th_pi  = theta * (1.0f / FLAME_PI);
    float str    = sinf(theta + r);
    float ctr    = cosf(theta - r);
    float spr    = sinf(FLAME_PI * r);
    float cpr    = cosf(FLAME_PI * r);

    float fx = x, fy = y, w = W.w[0];
    fx = (v == 1) ? sx                        : fx;
    fy = (v == 1) ? sy                        : fy;
    w  = (v == 1) ? W.w[1]                    : w;
    fx = (v == 2) ? x * inv_r2                : fx;
    fy = (v == 2) ? y * inv_r2                : fy;
    w  = (v == 2) ? W.w[2]                    : w;
    fx = (v == 3) ? fmaf(x, sr2, -y * cr2)    : fx;
    fy = (v == 3) ? fmaf(x, cr2,  y * sr2)    : fy;
    w  = (v == 3) ? W.w[3]                    : w;
    fx = (v == 4) ? (x - y) * (x + y) * inv_r : fx;
    fy = (v == 4) ? 2.0f * x * y * inv_r      : fy;
    w  = (v == 4) ? W.w[4]                    : w;
    fx = (v == 5) ? th_pi                     : fx;
    fy = (v == 5) ? (r - 1.0f)                : fy;
    w  = (v == 5) ? W.w[5]                    : w;
    fx = (v == 6) ? r * str                   : fx;
    fy = (v == 6) ? r * ctr                   : fy;
    w  = (v == 6) ? W.w[6]                    : w;
    fx = (v == 7) ? th_pi * spr               : fx;
    fy = (v == 7) ? th_pi * cpr               : fy;
    w  = (v == 7) ? W.w[7]                    : w;

    fx_o = fx * w;
    fy_o = fy * w;
    z_o  = z;
}

__global__ void __launch_bounds__(256) flame_kernel(
    const float* __restrict__ pts,   // N x 3
    const float* __restrict__ lin,   // 3 x 3
    const float* __restrict__ pst,   // 3 x 3
    const float* __restrict__ vps,   // 8
    const int*   __restrict__ chs,   // N
    float*       __restrict__ out,   // N x 3
    int n)
{
    const int lane  = threadIdx.x & 31;
    const int wslot = threadIdx.x >> 5;
    const int gwave = (int)((blockIdx.x * blockDim.x + threadIdx.x) >> 5);
    const int nwav  = (int)((gridDim.x * blockDim.x) >> 5);
    const int nfull = n & ~31;
    const int stride = nwav * 32;

    // Hoist all uniform small inputs into registers (s_load once, no VMEM in
    // the hot loop besides the async DMA and the output stores).
    Mat3 L; W8 W;
    #pragma unroll
    for (int i = 0; i < 9; ++i) L.m[i] = lin[i];
    #pragma unroll
    for (int i = 0; i < 8; ++i) W.w[i] = vps[i];

    // Per-wave double-buffered staging tiles, filled by async-to-LDS DMA:
    // 32 points * 12B = 384B (24 x b128, lanes 0-23) and 32 choices = 128B
    // (8 x b128, lanes 24-31). One async b128 per lane per chunk.
    __shared__ __align__(16) float ptile[WPB][2][96];
    __shared__ __align__(16) int   ctile[WPB][2][32];

    // Issue the async fill of one 32-point chunk into buffer `buf`.
    auto issue_chunk = [&](int base, int buf) {
        if (lane < 24) {
            const float* ga = pts + (size_t)base * 3 + lane * 4;
            unsigned lo = lds_off_u32(&ptile[wslot][buf][lane * 4]);
            asm volatile("global_load_async_to_lds_b128 %0, %1, off"
                         :: "v"(lo), "v"(ga) : "memory");
        } else {
            const int* gc = chs + base + (lane - 24) * 4;
            unsigned lo = lds_off_u32(&ctile[wslot][buf][(lane - 24) * 4]);
            asm volatile("global_load_async_to_lds_b128 %0, %1, off"
                         :: "v"(lo), "v"(gc) : "memory");
        }
    };

#if __has_builtin(__builtin_amdgcn_wmma_f32_16x16x4_f32)
    // B = post padded to 4x16 (K=4, N=16). Per ISA 16x16x4 f32 layout:
    // VGPR0: lanes 0-15 hold K=0 row, lanes 16-31 hold K=2 row;
    // VGPR1: lanes 0-15 hold K=1 row, lanes 16-31 hold K=3 row (zeros).
    const int bcol = lane & 15;
    float b0 = 0.0f, b1 = 0.0f;
    if (bcol < 3) {
        if (lane < 16) { b0 = pst[0 * 3 + bcol]; b1 = pst[1 * 3 + bcol]; }
        else           { b0 = pst[2 * 3 + bcol]; b1 = 0.0f; }
    }
    v2f B = { b0, b1 };
#endif

    // Prologue: prefetch this wave's first chunk.
    const int chunk0 = gwave * 32;
    if (chunk0 + 32 <= nfull) issue_chunk(chunk0, 0);

    int bi = 0;
    for (int base = chunk0; base + 32 <= nfull; base += stride, bi ^= 1) {
        const int nbase = base + stride;
        if (nbase + 32 <= nfull) {
            // WAR guard: prior reads of buffer bi^1 must have retired.
            asm volatile("s_wait_dscnt 0x0" ::: "memory");
            issue_chunk(nbase, bi ^ 1);
            // Wait for THIS chunk's 2 async ops; next chunk's 2 stay in flight.
            asm volatile("s_wait_asynccnt 0x2" ::: "memory");
        } else {
            asm volatile("s_wait_asynccnt 0x0" ::: "memory");
        }

        const float px = ptile[wslot][bi][lane * 3 + 0];
        const float py = ptile[wslot][bi][lane * 3 + 1];
        const float pz = ptile[wslot][bi][lane * 3 + 2];
        const int   v  = ctile[wslot][bi][lane];

        float fx, fy, z;
        flame_core(px, py, pz, v, L, W, fx, fy, z);

#if __has_builtin(__builtin_amdgcn_wmma_f32_16x16x4_f32)
        // ---- routed @ post via V_WMMA_F32_16X16X4_F32, 2 tiles of 16 points.
        // A (16x4): VGPR0 = {K0 = fx | K2 = z}, VGPR1 = {K1 = fy | K3 = 0}.
        // Tile 0 covers points base..base+15: lanes 16-31 need z of lane-16.
        float z_lo  = __shfl(z, lane & 15, 32);
        // Tile 1 covers points base+16..base+31: lanes 0-15 need fx/fy of lane+16.
        float fx_hi = __shfl(fx, (lane & 15) + 16, 32);
        float fy_hi = __shfl(fy, (lane & 15) + 16, 32);

        v2f A0 = { (lane < 16) ? fx    : z_lo, (lane < 16) ? fy    : 0.0f };
        v2f A1 = { (lane < 16) ? fx_hi : z,    (lane < 16) ? fy_hi : 0.0f };
        v8f C  = {};
        v8f d0 = __builtin_amdgcn_wmma_f32_16x16x4_f32(
            false, A0, false, B, (short)0, C, false, false);
        v8f d1 = __builtin_amdgcn_wmma_f32_16x16x4_f32(
            false, A1, false, B, (short)0, C, false, false);

        // D layout (16x16 f32): VGPR i -> rows M=i (lanes 0-15, N=lane) and
        // M=8+i (lanes 16-31, N=lane-16). Only N<3 columns are real outputs.
        // One base address per lane; all 16 stores use constant byte offsets
        // so they fold into a single s_clause of global_store_b32 offset:N.
        const int col = lane & 15;
        if (col < 3) {
            const int mb = (lane < 16) ? 0 : 8;
            float* __restrict__ o = out + (size_t)(base + mb) * 3 + col;
            #pragma unroll
            for (int i = 0; i < 8; ++i) {
                o[i * 3]      = d0[i];   // point base+mb+i
                o[48 + i * 3] = d1[i];   // point base+16+mb+i
            }
        }
#else
        const int p = base + lane;
        out[3 * (size_t)p + 0] = fmaf(fx, pst[0], fmaf(fy, pst[3], z * pst[6]));
        out[3 * (size_t)p + 1] = fmaf(fx, pst[1], fmaf(fy, pst[4], z * pst[7]));
        out[3 * (size_t)p + 2] = fmaf(fx, pst[2], fmaf(fy, pst[5], z * pst[8]));
#endif
    }

    // Tail (n % 32 != 0): plain scalar path, grid-stride over leftovers.
    for (int i = nfull + (int)(blockIdx.x * blockDim.x + threadIdx.x); i < n;
         i += (int)(gridDim.x * blockDim.x)) {
        float fx, fy, z;
        flame_core(pts[3 * (size_t)i + 0], pts[3 * (size_t)i + 1],
                   pts[3 * (size_t)i + 2], chs[i], L, W, fx, fy, z);
        out[3 * (size_t)i + 0] = fmaf(fx, pst[0], fmaf(fy, pst[3], z * pst[6]));
        out[3 * (size_t)i + 1] = fmaf(fx, pst[1], fmaf(fy, pst[4], z * pst[7]));
        out[3 * (size_t)i + 2] = fmaf(fx, pst[2], fmaf(fy, pst[5], z * pst[8]));
    }
}

extern "C" void kernel_launch(void* const* d_in, const int* in_sizes, int n_in,
                              void* d_out, int out_size, void* d_ws, size_t ws_size,
                              hipStream_t stream) {
    const float* pts = (const float*)d_in[0];  // points  (N*3)
    const float* lin = (const float*)d_in[1];  // linear  (9)
    const float* pst = (const float*)d_in[2];  // post    (9)
    const float* vps = (const float*)d_in[3];  // var_params (8)
    const int*   chs = (const int*)d_in[4];    // func_choices (N)
    float*       out = (float*)d_out;

    const int n = in_sizes[0] / 3;
    if (n <= 0) return;

    const int threads = 256;                   // 8 wave32 per block
    int blocks = (n + threads - 1) / threads;  // persistent-ish: cap at 4096
    if (blocks > 4096) blocks = 4096;          // -> ~4 chunks/wave at N=4M
    flame_kernel<<<blocks, threads, 0, stream>>>(pts, lin, pst, vps, chs, out, n);
}